// QueryAwarePooling_36962488549653
// MI455X (gfx1250) — compile-verified
//
#include <hip/hip_runtime.h>

#define N_NODES 50000
#define DIMV    256
#define BATCH   512

typedef __attribute__((ext_vector_type(2))) float v2f;
typedef __attribute__((ext_vector_type(8))) float v8f;

// ---------------------------------------------------------------------------
// K1: g_score[n] = dot(G[n,:], w_g)   (one wave32 per row, float4 loads)
// ---------------------------------------------------------------------------
__global__ void gscore_kernel(const float* __restrict__ G,
                              const float* __restrict__ w,   // attn_w[0:256]
                              float* __restrict__ gscore) {
  int lane   = threadIdx.x & 31;
  int wave   = (blockIdx.x * blockDim.x + threadIdx.x) >> 5;
  int nwaves = (gridDim.x * blockDim.x) >> 5;
  for (int n = wave; n < N_NODES; n += nwaves) {
    const float4* gp = (const float4*)(G + (size_t)n * DIMV + lane * 8);
    const float4* wp = (const float4*)(w + lane * 8);
    float4 g0 = gp[0], g1 = gp[1];
    float4 w0 = wp[0], w1 = wp[1];
    float s = g0.x*w0.x + g0.y*w0.y + g0.z*w0.z + g0.w*w0.w
            + g1.x*w1.x + g1.y*w1.y + g1.z*w1.z + g1.w*w1.w;
#pragma unroll
    for (int o = 16; o > 0; o >>= 1) s += __shfl_xor(s, o, 32);
    if (lane == 0) gscore[n] = s;
  }
}

// ---------------------------------------------------------------------------
// K2: single workgroup: gmax = max(g_score); e[n] = exp(g_score[n]-gmax);
//     Z = sum(e);   writes invZ and zeroes the accumulator.
//     (q_score[b] and bias cancel in softmax -> batch dim collapses)
// ---------------------------------------------------------------------------
__global__ void softmax_prep_kernel(const float* __restrict__ gscore,
                                    float* __restrict__ evals,
                                    float* __restrict__ accum,
                                    float* __restrict__ scalars) {
  __shared__ float red[1024];
  int t = threadIdx.x;

  float m = -3.402823466e38f;
  for (int i = t; i < N_NODES; i += 1024) m = fmaxf(m, gscore[i]);
  red[t] = m;
  __syncthreads();
  for (int o = 512; o > 0; o >>= 1) {
    if (t < o) red[t] = fmaxf(red[t], red[t + o]);
    __syncthreads();
  }
  float gmax = red[0];
  __syncthreads();

  float s = 0.0f;
  for (int i = t; i < N_NODES; i += 1024) {
    float ev = expf(gscore[i] - gmax);
    evals[i] = ev;
    s += ev;
  }
  red[t] = s;
  __syncthreads();
  for (int o = 512; o > 0; o >>= 1) {
    if (t < o) red[t] += red[t + o];
    __syncthreads();
  }
  if (t == 0) { scalars[0] = gmax; scalars[1] = 1.0f / red[0]; }
  if (t < DIMV) accum[t] = 0.0f;
}

// ---------------------------------------------------------------------------
// K3: pooled_row[d] += sum_n e[n] * G[n,d]  via V_WMMA_F32_16X16X4_F32.
//   16 nodes per iteration, 4 independent accumulator tiles so the XDL
//   accumulation chains interleave and loads batch into one clause.
//   A (16x4): lane L, vgpr v holds K = (L>>4)*2 + v (documented layout);
//             rows replicated -> all 16 tile rows identical.
//   B (4x16): assumed symmetric layout: vgpr v holds K=(L>>4)*2+v, N=L&15.
//   Row 0 of C lives in c[0] of lanes 0..15 -> LDS reduce -> global atomic.
// ---------------------------------------------------------------------------
#define WAVES_PER_BLOCK 4

__global__ void pooled_wmma_kernel(const float* __restrict__ G,
                                   const float* __restrict__ evals,
                                   float* __restrict__ accum) {
  __shared__ float sacc[16];
  const int d0    = blockIdx.x * 16;                 // 16-wide column chunk
  const int lane  = threadIdx.x & 31;
  const int wslot = threadIdx.x >> 5;
  const int wid   = blockIdx.y * WAVES_PER_BLOCK + wslot;
  const int nwN   = gridDim.y * WAVES_PER_BLOCK;     // waves along n
  const int half  = lane >> 4;                       // 0 or 1
  const int col   = lane & 15;
  const int d     = d0 + col;
  const int nstep = nwN * 16;                        // nodes per grid-stride

  if (threadIdx.x < 16) sacc[threadIdx.x] = 0.0f;
  __syncthreads();

  v8f c0 = {}, c1 = {}, c2 = {}, c3 = {};
  for (int n = wid * 16; n < N_NODES; n += nstep) {  // N_NODES % 16 == 0
    // speculative prefetch of next grid-stride tile (translation-fail safe)
    __builtin_prefetch(&G[(size_t)(n + nstep) * DIMV + d], 0, 1);

    const int k0 = n + half * 2;
    const int k1 = k0 + 4;
    const int k2 = k0 + 8;
    const int k3 = k0 + 12;

    v2f a0, a1, a2, a3, b0, b1, b2, b3;
    a0.x = evals[k0]; a0.y = evals[k0 + 1];
    a1.x = evals[k1]; a1.y = evals[k1 + 1];
    a2.x = evals[k2]; a2.y = evals[k2 + 1];
    a3.x = evals[k3]; a3.y = evals[k3 + 1];

    const float* g0p = G + (size_t)k0 * DIMV + d;
    const float* g1p = G + (size_t)k1 * DIMV + d;
    const float* g2p = G + (size_t)k2 * DIMV + d;
    const float* g3p = G + (size_t)k3 * DIMV + d;
    b0.x = g0p[0]; b0.y = g0p[DIMV];
    b1.x = g1p[0]; b1.y = g1p[DIMV];
    b2.x = g2p[0]; b2.y = g2p[DIMV];
    b3.x = g3p[0]; b3.y = g3p[DIMV];

    c0 = __builtin_amdgcn_wmma_f32_16x16x4_f32(false, a0, false, b0,
                                               (short)0, c0, false, false);
    c1 = __builtin_amdgcn_wmma_f32_16x16x4_f32(false, a1, false, b1,
                                               (short)0, c1, false, false);
    c2 = __builtin_amdgcn_wmma_f32_16x16x4_f32(false, a2, false, b2,
                                               (short)0, c2, false, false);
    c3 = __builtin_amdgcn_wmma_f32_16x16x4_f32(false, a3, false, b3,
                                               (short)0, c3, false, false);
  }

  // all 16 tile rows identical; lanes 0..15 hold row 0 cols d0..d0+15 in [0]
  float r = c0[0] + c1[0] + c2[0] + c3[0];
  if (lane < 16) atomicAdd(&sacc[col], r);           // ds_add_f32, per block
  __syncthreads();
  if (threadIdx.x < 16)                               // 1 global atomic / col
    atomicAdd(&accum[d0 + threadIdx.x], sacc[threadIdx.x]);
}

// ---------------------------------------------------------------------------
// K4: out[b, d] = accum[d] * invZ  for all 512 batch rows
// ---------------------------------------------------------------------------
__global__ void broadcast_kernel(const float* __restrict__ accum,
                                 const float* __restrict__ scalars,
                                 float* __restrict__ out) {
  float invZ = scalars[1];
  out[(size_t)blockIdx.x * DIMV + threadIdx.x] = accum[threadIdx.x] * invZ;
}

// ---------------------------------------------------------------------------
extern "C" void kernel_launch(void* const* d_in, const int* in_sizes, int n_in,
                              void* d_out, int out_size, void* d_ws, size_t ws_size,
                              hipStream_t stream) {
  (void)in_sizes; (void)n_in; (void)out_size; (void)ws_size;
  const float* G  = (const float*)d_in[0];   // graph_embs (50000 x 256)
  const float* aw = (const float*)d_in[2];   // attn_w (512); w_g = first 256
  float* out = (float*)d_out;                // (512 x 256)

  float* ws      = (float*)d_ws;
  float* gscore  = ws;                       // 50000
  float* evals   = ws + N_NODES;             // 50000
  float* accum   = ws + 2 * N_NODES;         // 256
  float* scalars = ws + 2 * N_NODES + DIMV;  // 2

  gscore_kernel<<<1024, 256, 0, stream>>>(G, aw, gscore);
  softmax_prep_kernel<<<1, 1024, 0, stream>>>(gscore, evals, accum, scalars);
  pooled_wmma_kernel<<<dim3(16, 64), 32 * WAVES_PER_BLOCK, 0, stream>>>(G, evals, accum);
  broadcast_kernel<<<BATCH, DIMV, 0, stream>>>(accum, scalars, out);
}